// GraphSAGEdge_74320114090101
// MI455X (gfx1250) — compile-verified
//
#include <hip/hip_runtime.h>
#include <hip/hip_bf16.h>

typedef float v2f __attribute__((ext_vector_type(2)));
typedef float v8f __attribute__((ext_vector_type(8)));

#define N_NODES  100000
#define F_IN     128
#define HID      16
#define C_OUT    8
#define N_EDGES  3200000

// ---------------------------------------------------------------------------
// Dense projection with native fp32 WMMA:  Yl = X @ Wl,  Yr = X @ Wr
// X: [nrows x K] row-major, W*: [K x 16] row-major, Y*: [nrows x 16].
// One wave32 per 16-row tile; V_WMMA_F32_16X16X4_F32 accumulates over K.
// ---------------------------------------------------------------------------
__global__ __launch_bounds__(256)
void sage_proj_wmma(const float* __restrict__ X,
                    const float* __restrict__ Wl,
                    const float* __restrict__ Wr,
                    float* __restrict__ Yl,
                    float* __restrict__ Yr,
                    int nrows, int K)
{
    const int lane   = threadIdx.x & 31;
    const int wave   = threadIdx.x >> 5;
    const int tile   = blockIdx.x * 8 + wave;
    const int ntiles = nrows >> 4;              // nrows is a multiple of 16
    if (tile >= ntiles) return;                 // wave-uniform: EXEC stays all-ones

    const int hi = lane >> 4;                   // half-wave selector
    const int m  = (tile << 4) + (lane & 15);   // A-fragment row for this lane
    const int n  = lane & 15;                   // B/C/D column for this lane

    v8f cl = {0.f, 0.f, 0.f, 0.f, 0.f, 0.f, 0.f, 0.f};
    v8f cr = cl;

    for (int kb = 0; kb < K; kb += 4) {
        // A 16x4: lane<16 -> K = kb+0,kb+1 ; lane>=16 -> K = kb+2,kb+3
        const float* ap = X + (long long)m * K + kb + 2 * hi;
        v2f a; a.x = ap[0]; a.y = ap[1];
        // B 4x16: VGPR0 -> K = kb+2*hi ; VGPR1 -> K = kb+1+2*hi ; col = n
        const int k0 = (kb + 2 * hi) * 16 + n;
        v2f bl; bl.x = Wl[k0]; bl.y = Wl[k0 + 16];
        v2f br; br.x = Wr[k0]; br.y = Wr[k0 + 16];

        cl = __builtin_amdgcn_wmma_f32_16x16x4_f32(false, a, false, bl,
                                                   (short)0, cl, false, false);
        cr = __builtin_amdgcn_wmma_f32_16x16x4_f32(false, a, false, br,
                                                   (short)0, cr, false, false);
    }

    // D 16x16: VGPR j -> row (j + 8*hi), col n
    const int rbase = (tile << 4) + 8 * hi;
#pragma unroll
    for (int j = 0; j < 8; ++j) {
        Yl[(long long)(rbase + j) * 16 + n] = cl[j];
        Yr[(long long)(rbase + j) * 16 + n] = cr[j];
    }
}

// ---------------------------------------------------------------------------
// Edge scatter: Acc[dst] += Y[src] (16 f32 atomics, L2-resident accumulator),
// optionally counting in-degree. One thread per edge.
// ---------------------------------------------------------------------------
__global__ __launch_bounds__(256)
void sage_scatter(const long long* __restrict__ ei,
                  const float* __restrict__ Y,
                  float* __restrict__ Acc,
                  float* __restrict__ cnt,
                  int E)
{
    int e = blockIdx.x * blockDim.x + threadIdx.x;
    if (e >= E) return;
    const int s = (int)ei[e];
    const int d = (int)ei[(long long)E + e];
    if (cnt) atomicAdd(&cnt[d], 1.0f);

    const float4* ys = (const float4*)(Y + (long long)s * 16);
    float4 v0 = ys[0], v1 = ys[1], v2 = ys[2], v3 = ys[3];
    float* o = Acc + (long long)d * 16;
    atomicAdd(o + 0,  v0.x); atomicAdd(o + 1,  v0.y);
    atomicAdd(o + 2,  v0.z); atomicAdd(o + 3,  v0.w);
    atomicAdd(o + 4,  v1.x); atomicAdd(o + 5,  v1.y);
    atomicAdd(o + 6,  v1.z); atomicAdd(o + 7,  v1.w);
    atomicAdd(o + 8,  v2.x); atomicAdd(o + 9,  v2.y);
    atomicAdd(o + 10, v2.z); atomicAdd(o + 11, v2.w);
    atomicAdd(o + 12, v3.x); atomicAdd(o + 13, v3.y);
    atomicAdd(o + 14, v3.z); atomicAdd(o + 15, v3.w);
}

// ---------------------------------------------------------------------------
// h = relu(Acc / max(cnt,1) + bias + Yr), written in place into Acc.
// ---------------------------------------------------------------------------
__global__ __launch_bounds__(256)
void sage_finish(float* __restrict__ Acc,
                 const float* __restrict__ Yr,
                 const float* __restrict__ bias,
                 const float* __restrict__ cnt,
                 int N)
{
    int i = blockIdx.x * blockDim.x + threadIdx.x;
    if (i >= N * 16) return;
    const int node = i >> 4, k = i & 15;
    float c = cnt[node]; c = (c > 1.0f) ? c : 1.0f;
    float v = Acc[i] / c + bias[k] + Yr[i];
    Acc[i] = (v > 0.0f) ? v : 0.0f;
}

// ---------------------------------------------------------------------------
// Edge head: logits = concat(h[src], h[dst]) @ Wc + bc ; log_softmax over 8.
// One thread per edge; Wc/bc staged in LDS; gathers hit L2 (h = 6.4 MB).
// ---------------------------------------------------------------------------
__global__ __launch_bounds__(256)
void edge_head(const long long* __restrict__ ei,
               const float* __restrict__ h,
               const float* __restrict__ Wc,
               const float* __restrict__ bc,
               float* __restrict__ out,
               int E)
{
    __shared__ float sW[32 * 8];
    __shared__ float sb[8];
    for (int i = threadIdx.x; i < 256; i += blockDim.x) sW[i] = Wc[i];
    if (threadIdx.x < 8) sb[threadIdx.x] = bc[threadIdx.x];
    __syncthreads();

    int e = blockIdx.x * blockDim.x + threadIdx.x;
    if (e >= E) return;
    const int s = (int)ei[e];
    const int d = (int)ei[(long long)E + e];

    float f[32];
    const float4* hs = (const float4*)(h + (long long)s * 16);
    const float4* hd = (const float4*)(h + (long long)d * 16);
#pragma unroll
    for (int q = 0; q < 4; ++q) {
        float4 v = hs[q];
        f[q * 4 + 0] = v.x; f[q * 4 + 1] = v.y;
        f[q * 4 + 2] = v.z; f[q * 4 + 3] = v.w;
    }
#pragma unroll
    for (int q = 0; q < 4; ++q) {
        float4 v = hd[q];
        f[16 + q * 4 + 0] = v.x; f[16 + q * 4 + 1] = v.y;
        f[16 + q * 4 + 2] = v.z; f[16 + q * 4 + 3] = v.w;
    }

    float logits[8];
#pragma unroll
    for (int c = 0; c < 8; ++c) {
        float acc = sb[c];
#pragma unroll
        for (int k = 0; k < 32; ++k) acc = fmaf(f[k], sW[k * 8 + c], acc);
        logits[c] = acc;
    }

    float m = logits[0];
#pragma unroll
    for (int c = 1; c < 8; ++c) m = fmaxf(m, logits[c]);
    float sum = 0.0f;
#pragma unroll
    for (int c = 0; c < 8; ++c) { logits[c] -= m; sum += __expf(logits[c]); }
    const float lse = __logf(sum);

    float4 o0 = make_float4(logits[0] - lse, logits[1] - lse,
                            logits[2] - lse, logits[3] - lse);
    float4 o1 = make_float4(logits[4] - lse, logits[5] - lse,
                            logits[6] - lse, logits[7] - lse);
    float4* op = (float4*)(out + (long long)e * 8);
    op[0] = o0; op[1] = o1;
}

// ---------------------------------------------------------------------------
extern "C" void kernel_launch(void* const* d_in, const int* in_sizes, int n_in,
                              void* d_out, int out_size, void* d_ws, size_t ws_size,
                              hipStream_t stream)
{
    const float*     x   = (const float*)d_in[0];
    const long long* ei  = (const long long*)d_in[1];   // int64 per reference
    const float*     W1l = (const float*)d_in[2];
    const float*     b1  = (const float*)d_in[3];
    const float*     W1r = (const float*)d_in[4];
    const float*     W2l = (const float*)d_in[5];
    const float*     b2  = (const float*)d_in[6];
    const float*     W2r = (const float*)d_in[7];
    const float*     Wc  = (const float*)d_in[8];
    const float*     bc  = (const float*)d_in[9];
    float*           out = (float*)d_out;

    const int N = N_NODES, E = N_EDGES;
    const size_t NB = (size_t)N * 16;

    float* ws  = (float*)d_ws;
    float* A   = ws;            // y1l, later y2l   [N x 16]
    float* B   = ws + NB;       // y1r, later y2r   [N x 16]
    float* Cg  = ws + 2 * NB;   // agg1 -> h1       [N x 16]
    float* Eg  = ws + 3 * NB;   // agg2 -> h2       [N x 16]
    float* cnt = ws + 4 * NB;   // in-degree        [N]

    // zero accumulators + counts every call (deterministic; Cg,Eg,cnt contiguous)
    hipMemsetAsync(Cg, 0, (2 * NB + (size_t)N) * sizeof(float), stream);

    const int tiles   = N / 16;            // 6250 exact
    const int pblocks = (tiles + 7) / 8;   // 8 waves/block
    const int eblocks = (E + 255) / 256;
    const int fblocks = (N * 16 + 255) / 256;

    // Layer 1 (project in 128-dim once, aggregate in 16-dim)
    sage_proj_wmma<<<pblocks, 256, 0, stream>>>(x, W1l, W1r, A, B, N, F_IN);
    sage_scatter  <<<eblocks, 256, 0, stream>>>(ei, A, Cg, cnt, E);
    sage_finish   <<<fblocks, 256, 0, stream>>>(Cg, B, b1, cnt, N);

    // Layer 2
    sage_proj_wmma<<<pblocks, 256, 0, stream>>>(Cg, W2l, W2r, A, B, N, HID);
    sage_scatter  <<<eblocks, 256, 0, stream>>>(ei, A, Eg, nullptr, E);
    sage_finish   <<<fblocks, 256, 0, stream>>>(Eg, B, b2, cnt, N);

    // Edge classification head
    edge_head     <<<eblocks, 256, 0, stream>>>(ei, Eg, Wc, bc, out, E);
}